// SNG_38036230373377
// MI455X (gfx1250) — compile-verified
//
#include <hip/hip_runtime.h>
#include <stdint.h>

// ---- problem constants (from reference) ----
#define BATCH 1024
#define PBITS 8
#define LBITS 16384
#define TLEN  (PBITS + LBITS)   // 16392 columns per (b,i) row of r

// ---- tiling ----
#define NL      256              // output bits per tile per wave
#define ROWW    (NL + PBITS)     // 264 dwords per LDS tile row (covers +i shift)
#define WAVES   4                // waves per block
#define TILES   4                // tiles per wave
#define LCHUNK  (WAVES * TILES * NL)   // 4096 output bits per block

typedef __attribute__((ext_vector_type(4))) unsigned int u32x4;
typedef __attribute__((ext_vector_type(8))) int          i32x8;
typedef __attribute__((ext_vector_type(4))) int          i32x4;

__global__ __launch_bounds__(WAVES * 32)
void SNG_tdm_kernel(const int* __restrict__ num,
                    const int* __restrict__ r,
                    float* __restrict__ out) {
    // per-wave double-buffered tiles: [wave][buf][row i][col]
    __shared__ uint32_t smem[WAVES * 2 * PBITS * ROWW];

    const int wave = threadIdx.x >> 5;
    const int lane = threadIdx.x & 31;
    const int b    = blockIdx.y;
    const int l_wave = blockIdx.x * LCHUNK + wave * (TILES * NL);

    uint32_t* wbuf = &smem[wave * 2 * PBITS * ROWW];
    const uint32_t lds0 = (uint32_t)(uintptr_t)(&wbuf[0]);
    const uint32_t lds1 = (uint32_t)(uintptr_t)(&wbuf[PBITS * ROWW]);

    // global byte address of row (b, i=0, col 0)
    const uint64_t rbase = (uint64_t)(uintptr_t)r +
                           (uint64_t)b * PBITS * TLEN * 4ull;

    const uint32_t x = (uint32_t)num[b];   // block-uniform -> scalar load

    // Issue one TDM load: 8 rows x ROWW dwords, row stride TLEN dwords,
    // packed contiguously into LDS at lds_addr.
    auto issue_tdm = [&](int tile, uint32_t lds_addr) {
        const uint64_t ga = rbase + (uint64_t)(l_wave + tile * NL) * 4ull;
        u32x4 g0;
        g0.x = 1u;                                   // count=1 (valid), user mode
        g0.y = lds_addr;                             // LDS byte address
        g0.z = (uint32_t)(ga & 0xFFFFFFFFu);         // global_addr[31:0]
        g0.w = (uint32_t)((ga >> 32) & 0x01FFFFFFu)  // global_addr[56:32]
             | 0x80000000u;                          // type = 2 ("image")
        i32x8 g1;
        g1[0] = 0x00020000;                          // data_size=2 (4B), no mask/pad
        g1[1] = (int)((TLEN & 0xFFFF) << 16);        // tensor_dim0[15:0]=16392
        g1[2] = (int)(((TLEN >> 16) & 0xFFFF)        // tensor_dim0[31:16]
              | (PBITS << 16));                      // tensor_dim1[15:0]=8
        g1[3] = (int)(ROWW << 16);                   // tile_dim0=264 (td1 hi=0)
        g1[4] = PBITS;                               // tile_dim1=8, tile_dim2=0
        g1[5] = TLEN;                                // tensor_dim0_stride[31:0]
        g1[6] = 0;                                   // stride hi / dim1_stride lo
        g1[7] = 0;
        i32x4 g2 = {0, 0, 0, 0};                     // unused (2D tile)
        i32x4 g3 = {0, 0, 0, 0};
        i32x8 g4 = {0, 0, 0, 0, 0, 0, 0, 0};        // trailing words (unused)
        __builtin_amdgcn_tensor_load_to_lds(g0, g1, g2, g3, g4, 0);
    };

    issue_tdm(0, lds0);
    issue_tdm(1, lds1);

    for (int t = 0; t < TILES; ++t) {
        // Oldest outstanding TDM is tile t (in-order per wave).
        if (t + 1 < TILES) __builtin_amdgcn_s_wait_tensorcnt(1);
        else               __builtin_amdgcn_s_wait_tensorcnt(0);

        const uint32_t* buf = &wbuf[(t & 1) * PBITS * ROWW];
        const int ltile = l_wave + t * NL;

#pragma unroll
        for (int q = 0; q < NL / 32; ++q) {
            const int j = lane + q * 32;
            uint32_t y = 0u;
#pragma unroll
            for (int i = 0; i < PBITS; ++i) {
                const uint32_t rv = buf[i * ROWW + j + i]; // conflict-free ds_load
                const uint32_t xi = (x >> i) & 1u;
                // maj(x,y,r): x ? (y|r) : (y&r)   (values live in bit 0)
                y = (y & rv) | (xi & (y | rv));
            }
            const float fo = (float)y;
            __builtin_nontemporal_store(fo, &out[(size_t)b * LBITS + ltile + j]);
        }

        // Ensure all LDS reads of this buffer completed before TDM overwrites it.
        asm volatile("s_wait_dscnt 0x0" ::: "memory");
        if (t + 2 < TILES) issue_tdm(t + 2, (t & 1) ? lds1 : lds0);
    }
}

extern "C" void kernel_launch(void* const* d_in, const int* in_sizes, int n_in,
                              void* d_out, int out_size, void* d_ws, size_t ws_size,
                              hipStream_t stream) {
    const int* num = (const int*)d_in[0];   // [B] int32
    const int* r   = (const int*)d_in[1];   // [B, P, P+L] int32 bits
    float* out     = (float*)d_out;         // [B, L]

    dim3 grid(LBITS / LCHUNK, BATCH, 1);    // (4, 1024)
    dim3 block(WAVES * 32, 1, 1);           // 128 threads = 4 waves
    SNG_tdm_kernel<<<grid, block, 0, stream>>>(num, r, out);
}